// SACN_6854767804918
// MI455X (gfx1250) — compile-verified
//
#include <hip/hip_runtime.h>
#include <hip/hip_bf16.h>
#include <math.h>

// Problem constants (from reference)
#define N_ENT   200000
#define N_RELS  500
#define E_EDGE  500000
#define BATCH   256
#define INIT_F  100
#define GC1_F   150
#define D_F     200
#define C_CH    200
#define K_W     5
#define EPSV    1e-5f

typedef float v2f __attribute__((ext_vector_type(2)));
typedef float v8f __attribute__((ext_vector_type(8)));

__device__ __forceinline__ void atomAdd(float* p, float v) {
  __hip_atomic_fetch_add(p, v, __ATOMIC_RELAXED, __HIP_MEMORY_SCOPE_AGENT);
}

// ---------------- elementwise helpers ----------------
__global__ void k_zero(float* p, long n) {
  long i = blockIdx.x * (long)blockDim.x + threadIdx.x;
  if (i < n) p[i] = 0.f;
}

// O[r, f] = bias[f] for f < F, 0 in pad columns (row stride ld)
__global__ void k_init_bias(float* O, const float* __restrict__ bias,
                            int N, int ld, int F) {
  long i = blockIdx.x * (long)blockDim.x + threadIdx.x;
  long total = (long)N * ld;
  if (i >= total) return;
  int f = (int)(i % ld);
  O[i] = (f < F) ? bias[f] : 0.f;
}

// ---------------- WMMA f32 GEMM: C[M x ldC] (+)= A[M x ldA] * W[Kw x ldW] ----------------
// One wave per 16x16 C tile. grid = (ceil(Ncols/16), M/16, Ksplit), blockDim = 32.
// A must be zero-padded in columns [Kw, Kchunk*gridZ) if that range is iterated.
template<bool ATOMIC>
__global__ void k_gemm_w(const float* __restrict__ A, int ldA,
                         const float* __restrict__ W, int ldW, int Kw,
                         float* __restrict__ C, int ldC, int Ncols,
                         int Kchunk) {
  const int lane = threadIdx.x & 31;
  const int l15  = lane & 15;
  const int hf   = lane >> 4;            // lane half: 0 or 1
  const int n0 = blockIdx.x * 16;
  const int m0 = blockIdx.y * 16;
  const long arow = m0 + l15;            // A-matrix: lane holds one M row
  const int  bcol = n0 + l15;            // B-matrix: lane holds one N col
  const bool bvalid = (bcol < Ncols);
  const int k0 = blockIdx.z * Kchunk;
  v8f c = {0.f,0.f,0.f,0.f,0.f,0.f,0.f,0.f};
  for (int k = k0; k < k0 + Kchunk; k += 4) {
    const int ka = k + hf * 2;           // A 16x4 f32 layout: v0:K={0,2}, v1:K={1,3}
    v2f a, b;
    a.x = A[arow * ldA + ka];
    a.y = A[arow * ldA + ka + 1];
    b.x = (bvalid && (ka     < Kw)) ? W[(long)ka       * ldW + bcol] : 0.f;
    b.y = (bvalid && (ka + 1 < Kw)) ? W[(long)(ka + 1) * ldW + bcol] : 0.f;
    c = __builtin_amdgcn_wmma_f32_16x16x4_f32(false, a, false, b, (short)0, c,
                                              false, false);
  }
  if (bvalid) {
#pragma unroll
    for (int r = 0; r < 8; ++r) {
      const long row = m0 + r + 8 * hf;  // C/D layout: VGPR r -> M = r + 8*half
      if (ATOMIC) atomAdd(&C[row * ldC + bcol], c[r]);
      else        C[row * ldC + bcol] = c[r];
    }
  }
}

// ---------------- final scores: out = sigmoid(Z[256x200] @ Eall[200000x200]^T) ----------------
__global__ void k_score(const float* __restrict__ Z,
                        const float* __restrict__ Eall,
                        float* __restrict__ out) {
  const int lane = threadIdx.x & 31;
  const int l15 = lane & 15;
  const int hf  = lane >> 4;
  const int n0 = blockIdx.x * 16;        // entity tile
  const int m0 = blockIdx.y * 16;        // batch tile
  const long arow = m0 + l15;
  const long ecol = n0 + l15;            // entity index for this lane's B column
  const float* erow = Eall + ecol * D_F;
  v8f c = {0.f,0.f,0.f,0.f,0.f,0.f,0.f,0.f};
  for (int k = 0; k < D_F; k += 4) {
    const int ka = k + hf * 2;
    v2f a, b;
    a.x = Z[arow * D_F + ka];
    a.y = Z[arow * D_F + ka + 1];
    b.x = erow[ka];                      // B[k][n] = Eall[n][k] (transposed access)
    b.y = erow[ka + 1];
    c = __builtin_amdgcn_wmma_f32_16x16x4_f32(false, a, false, b, (short)0, c,
                                              false, false);
  }
#pragma unroll
  for (int r = 0; r < 8; ++r) {
    const long row = m0 + r + 8 * hf;
    float v = c[r];
    out[row * (long)N_ENT + n0 + l15] = 1.f / (1.f + expf(-v));
  }
}

// ---------------- GCN scatter: O[row] += a*S[col]; O[col] += a*S[row] ----------------
__global__ void k_scatter(const float* __restrict__ S, int ld,
                          float* __restrict__ O,
                          const int* __restrict__ rows,
                          const int* __restrict__ cols,
                          const int* __restrict__ rel,
                          const float* __restrict__ alpha, int F) {
  const int e = blockIdx.x;
  const int f = threadIdx.x;
  if (f >= F) return;
  const int r = rows[e];
  const int c = cols[e];
  const float alp = alpha[rel[e]];       // alpha[(N_REL+1),1], rel in [0,N_REL)
  const float sc = S[(long)c * ld + f];
  const float sr = S[(long)r * ld + f];
  atomAdd(&O[(long)r * ld + f], alp * sc);
  atomAdd(&O[(long)c * ld + f], alp * sr);
}

// ---------------- per-column stats over N rows (atomic partials) ----------------
__global__ void k_colstats(const float* __restrict__ X, int ld, int N, int F,
                           float* __restrict__ sum, float* __restrict__ sumsq) {
  const int f = threadIdx.x;
  if (f >= F) return;
  const int chunk = (N + gridDim.x - 1) / gridDim.x;
  const int r0 = blockIdx.x * chunk;
  const int r1 = (r0 + chunk < N) ? (r0 + chunk) : N;
  float s = 0.f, q = 0.f;
  for (int r = r0; r < r1; ++r) {
    float v = X[(long)r * ld + f];
    s += v; q += v * v;
  }
  atomAdd(&sum[f], s);
  atomAdd(&sumsq[f], q);
}

__global__ void k_finstats(const float* __restrict__ sum,
                           const float* __restrict__ sumsq,
                           int N, int F,
                           float* __restrict__ mean, float* __restrict__ rstd) {
  const int f = threadIdx.x;
  if (f >= F) return;
  const float m = sum[f] / (float)N;
  const float v = sumsq[f] / (float)N - m * m;
  mean[f] = m;
  rstd[f] = rsqrtf(v + EPSV);
}

// Y[r,f] = tanh((X[r,f]-m[f])*rs[f]*g[f]+b[f]) for f<F; 0 in pads of Y (row strides ldX/ldY)
__global__ void k_bn_tanh(const float* __restrict__ X, int ldX,
                          float* __restrict__ Y, int ldY,
                          const float* __restrict__ mean,
                          const float* __restrict__ rstd,
                          const float* __restrict__ g,
                          const float* __restrict__ b, int N, int F) {
  long i = blockIdx.x * (long)blockDim.x + threadIdx.x;
  long total = (long)N * ldY;
  if (i >= total) return;
  int f = (int)(i % ldY);
  long r = i / ldY;
  float v = 0.f;
  if (f < F) v = tanhf((X[r * ldX + f] - mean[f]) * rstd[f] * g[f] + b[f]);
  Y[i] = v;
}

// ---------------- batch head ----------------
__global__ void k_gather(const float* __restrict__ Eall,
                         const float* __restrict__ emb_rel,
                         const int* __restrict__ e1,
                         const int* __restrict__ rel,
                         float* __restrict__ st) {  // [B,2,D]
  int i = blockIdx.x * blockDim.x + threadIdx.x;
  if (i >= BATCH * 2 * D_F) return;
  int b = i / (2 * D_F);
  int r = i % (2 * D_F);
  int c = r / D_F;
  int d = r % D_F;
  st[i] = (c == 0) ? Eall[(long)e1[b] * D_F + d]
                   : emb_rel[(long)rel[b] * D_F + d];
}

// BN0 stats: per channel (2) over B*D elements. grid=2, block=256.
__global__ void k_ch2stats(const float* __restrict__ st,
                           float* __restrict__ mean, float* __restrict__ rstd) {
  const int c = blockIdx.x;
  float s = 0.f, q = 0.f;
  for (int i = threadIdx.x; i < BATCH * D_F; i += blockDim.x) {
    int b = i / D_F, d = i % D_F;
    float v = st[b * 2 * D_F + c * D_F + d];
    s += v; q += v * v;
  }
  __shared__ float ss[256], qq[256];
  ss[threadIdx.x] = s; qq[threadIdx.x] = q;
  __syncthreads();
  for (int o = 128; o > 0; o >>= 1) {
    if ((int)threadIdx.x < o) { ss[threadIdx.x] += ss[threadIdx.x + o];
                                qq[threadIdx.x] += qq[threadIdx.x + o]; }
    __syncthreads();
  }
  if (threadIdx.x == 0) {
    float m = ss[0] / (float)(BATCH * D_F);
    float v = qq[0] / (float)(BATCH * D_F) - m * m;
    mean[c] = m; rstd[c] = rsqrtf(v + EPSV);
  }
}

// conv1d, pad=2: xc[b,co,d] = cb[co] + sum_ci sum_k w[co,ci,k]*bn0(st[b,ci,d+k-2])
__global__ void k_conv(const float* __restrict__ st,
                       const float* __restrict__ m0, const float* __restrict__ r0,
                       const float* __restrict__ g0, const float* __restrict__ b0,
                       const float* __restrict__ w,  const float* __restrict__ cb,
                       float* __restrict__ xc) {
  __shared__ float s0[D_F + 4], s1[D_F + 4], sw[C_CH * 2 * K_W];
  const int b = blockIdx.x;
  for (int i = threadIdx.x; i < D_F + 4; i += blockDim.x) {
    int d = i - 2;
    bool in = (d >= 0 && d < D_F);
    s0[i] = in ? (st[b * 2 * D_F + d]       - m0[0]) * r0[0] * g0[0] + b0[0] : 0.f;
    s1[i] = in ? (st[b * 2 * D_F + D_F + d] - m0[1]) * r0[1] * g0[1] + b0[1] : 0.f;
  }
  for (int i = threadIdx.x; i < C_CH * 2 * K_W; i += blockDim.x) sw[i] = w[i];
  __syncthreads();
  for (int co = 0; co < C_CH; ++co) {
    const float* w0 = &sw[co * 2 * K_W];
    for (int d = threadIdx.x; d < D_F; d += blockDim.x) {
      float acc = cb[co];
#pragma unroll
      for (int k = 0; k < K_W; ++k)
        acc += w0[k] * s0[d + k] + w0[K_W + k] * s1[d + k];
      xc[((long)b * C_CH + co) * D_F + d] = acc;
    }
  }
}

// BN1 stats: per co over [b,d]. grid=C_CH, block=256.
__global__ void k_chanstats(const float* __restrict__ xc,
                            float* __restrict__ mean, float* __restrict__ rstd) {
  const int co = blockIdx.x;
  float s = 0.f, q = 0.f;
  for (int i = threadIdx.x; i < BATCH * D_F; i += blockDim.x) {
    int b = i / D_F, d = i % D_F;
    float v = xc[((long)b * C_CH + co) * D_F + d];
    s += v; q += v * v;
  }
  __shared__ float ss[256], qq[256];
  ss[threadIdx.x] = s; qq[threadIdx.x] = q;
  __syncthreads();
  for (int o = 128; o > 0; o >>= 1) {
    if ((int)threadIdx.x < o) { ss[threadIdx.x] += ss[threadIdx.x + o];
                                qq[threadIdx.x] += qq[threadIdx.x + o]; }
    __syncthreads();
  }
  if (threadIdx.x == 0) {
    float m = ss[0] / (float)(BATCH * D_F);
    float v = qq[0] / (float)(BATCH * D_F) - m * m;
    mean[co] = m; rstd[co] = rsqrtf(v + EPSV);
  }
}

// xn[i] = relu(bn1(xc[i]))  (layout [B, C, D] flattens to fc's K = co*D + d)
__global__ void k_bnrelu(const float* __restrict__ xc,
                         const float* __restrict__ m1, const float* __restrict__ r1,
                         const float* __restrict__ g1, const float* __restrict__ b1,
                         float* __restrict__ xn, long n) {
  long i = blockIdx.x * (long)blockDim.x + threadIdx.x;
  if (i >= n) return;
  int co = (int)((i / D_F) % C_CH);
  float v = (xc[i] - m1[co]) * r1[co] * g1[co] + b1[co];
  xn[i] = v > 0.f ? v : 0.f;
}

// BN2 + relu on z[256x200] -> z2 (single small block; stats fused)
__global__ void k_bn2(const float* __restrict__ z,
                      const float* __restrict__ g, const float* __restrict__ b,
                      float* __restrict__ z2) {
  const int c = threadIdx.x;
  if (c >= D_F) return;
  float s = 0.f, q = 0.f;
  for (int r = 0; r < BATCH; ++r) {
    float v = z[r * D_F + c];
    s += v; q += v * v;
  }
  float m = s / (float)BATCH;
  float va = q / (float)BATCH - m * m;
  float rs = rsqrtf(va + EPSV);
  for (int r = 0; r < BATCH; ++r) {
    float v = (z[r * D_F + c] - m) * rs * g[c] + b[c];
    z2[r * D_F + c] = v > 0.f ? v : 0.f;
  }
}

// ---------------- host launch ----------------
extern "C" void kernel_launch(void* const* d_in, const int* in_sizes, int n_in,
                              void* d_out, int out_size, void* d_ws, size_t ws_size,
                              hipStream_t stream) {
  const float* emb_e   = (const float*)d_in[0];
  const float* gc1_w   = (const float*)d_in[1];
  const float* gc1_b   = (const float*)d_in[2];
  const float* alpha1  = (const float*)d_in[3];
  const float* gc2_w   = (const float*)d_in[4];
  const float* gc2_b   = (const float*)d_in[5];
  const float* alpha2  = (const float*)d_in[6];
  const float* emb_rel = (const float*)d_in[7];
  const float* conv_w  = (const float*)d_in[8];
  const float* conv_b  = (const float*)d_in[9];
  const float* fc_w    = (const float*)d_in[10];
  const float* fc_b    = (const float*)d_in[11];
  const float* bn0_g   = (const float*)d_in[12];
  const float* bn0_b   = (const float*)d_in[13];
  const float* bn1_g   = (const float*)d_in[14];
  const float* bn1_b   = (const float*)d_in[15];
  const float* bn2_g   = (const float*)d_in[16];
  const float* bn2_b   = (const float*)d_in[17];
  const float* bn3_g   = (const float*)d_in[18];
  const float* bn3_b   = (const float*)d_in[19];
  const float* bn4_g   = (const float*)d_in[20];
  const float* bn4_b   = (const float*)d_in[21];
  const int*   e1      = (const int*)d_in[22];
  const int*   rel     = (const int*)d_in[23];
  /* d_in[24] = X (arange) unused */
  const int*   adj_idx = (const int*)d_in[25];
  const int*   adj_rel = (const int*)d_in[26];
  const int* rows = adj_idx;
  const int* cols = adj_idx + E_EDGE;
  float* out = (float*)d_out;

  // workspace carve (~576 MB total)
  char* ws = (char*)d_ws;
  float* s1 = (float*)(ws);                         // 200000*160 f32 -> later h1
  float* x1 = (float*)(ws + 128000000LL);           // 200000*160 f32 -> later head bufs
  float* h1 = s1;
  float* s2 = (float*)(ws + 256000000LL);           // 200000*200 f32 -> later e_all
  float* x2 = (float*)(ws + 416000000LL);           // 200000*200 f32
  float* e_all = s2;
  float* st = x1;                                            // [256,2,200]
  float* xc = (float*)(ws + 128000000LL + 1048576LL);        // [256,200,200]
  float* xn = (float*)(ws + 128000000LL + 1048576LL + 41943040LL);
  float* z  = (float*)(ws + 128000000LL + 1048576LL + 83886080LL);
  float* z2 = (float*)(ws + 128000000LL + 1048576LL + 83886080LL + 262144LL);
  float* stats = (float*)(ws + 576000000LL);
  float* SUM = stats, *SQ = stats + 256, *MEAN = stats + 512, *RSTD = stats + 768;
  float* M0 = stats + 1024; float* R0 = stats + 1032;
  float* M1 = stats + 1040; float* R1 = stats + 1296;

  // ---- GCN layer 1 ----
  {
    long tot = (long)N_ENT * 160;
    k_init_bias<<<dim3((unsigned)((tot + 255) / 256)), dim3(256), 0, stream>>>(
        x1, gc1_b, N_ENT, 160, GC1_F);
  }
  k_gemm_w<false><<<dim3(10, N_ENT / 16, 1), dim3(32), 0, stream>>>(
      emb_e, INIT_F, gc1_w, GC1_F, INIT_F, s1, 160, GC1_F, INIT_F);
  k_scatter<<<dim3(E_EDGE), dim3(256), 0, stream>>>(
      s1, 160, x1, rows, cols, adj_rel, alpha1, GC1_F);
  k_zero<<<dim3(2), dim3(256), 0, stream>>>(stats, 512);
  k_colstats<<<dim3(256), dim3(256), 0, stream>>>(x1, 160, N_ENT, GC1_F, SUM, SQ);
  k_finstats<<<dim3(1), dim3(256), 0, stream>>>(SUM, SQ, N_ENT, GC1_F, MEAN, RSTD);
  {
    long tot = (long)N_ENT * 160;
    k_bn_tanh<<<dim3((unsigned)((tot + 255) / 256)), dim3(256), 0, stream>>>(
        x1, 160, h1, 160, MEAN, RSTD, bn3_g, bn3_b, N_ENT, GC1_F);
  }

  // ---- GCN layer 2 ----
  {
    long tot = (long)N_ENT * D_F;
    k_init_bias<<<dim3((unsigned)((tot + 255) / 256)), dim3(256), 0, stream>>>(
        x2, gc2_b, N_ENT, D_F, D_F);
  }
  k_gemm_w<false><<<dim3(13, N_ENT / 16, 1), dim3(32), 0, stream>>>(
      h1, 160, gc2_w, D_F, GC1_F, s2, D_F, D_F, 160);
  k_scatter<<<dim3(E_EDGE), dim3(256), 0, stream>>>(
      s2, D_F, x2, rows, cols, adj_rel, alpha2, D_F);
  k_zero<<<dim3(2), dim3(256), 0, stream>>>(stats, 512);
  k_colstats<<<dim3(256), dim3(256), 0, stream>>>(x2, D_F, N_ENT, D_F, SUM, SQ);
  k_finstats<<<dim3(1), dim3(256), 0, stream>>>(SUM, SQ, N_ENT, D_F, MEAN, RSTD);
  {
    long tot = (long)N_ENT * D_F;
    k_bn_tanh<<<dim3((unsigned)((tot + 255) / 256)), dim3(256), 0, stream>>>(
        x2, D_F, e_all, D_F, MEAN, RSTD, bn4_g, bn4_b, N_ENT, D_F);
  }

  // ---- batch head ----
  k_gather<<<dim3((BATCH * 2 * D_F + 255) / 256), dim3(256), 0, stream>>>(
      e_all, emb_rel, e1, rel, st);
  k_ch2stats<<<dim3(2), dim3(256), 0, stream>>>(st, M0, R0);
  k_conv<<<dim3(BATCH), dim3(256), 0, stream>>>(
      st, M0, R0, bn0_g, bn0_b, conv_w, conv_b, xc);
  k_chanstats<<<dim3(C_CH), dim3(256), 0, stream>>>(xc, M1, R1);
  {
    long tot = (long)BATCH * C_CH * D_F;
    k_bnrelu<<<dim3((unsigned)((tot + 255) / 256)), dim3(256), 0, stream>>>(
        xc, M1, R1, bn1_g, bn1_b, xn, tot);
  }
  k_init_bias<<<dim3((BATCH * D_F + 255) / 256), dim3(256), 0, stream>>>(
      z, fc_b, BATCH, D_F, D_F);
  // fc GEMM, K=40000 split across 40 chunks with atomic accumulation
  k_gemm_w<true><<<dim3(13, BATCH / 16, 40), dim3(32), 0, stream>>>(
      xn, C_CH * D_F, fc_w, D_F, C_CH * D_F, z, D_F, D_F, 1000);
  k_bn2<<<dim3(1), dim3(256), 0, stream>>>(z, bn2_g, bn2_b, z2);

  // ---- final scores with fused sigmoid ----
  k_score<<<dim3(N_ENT / 16, BATCH / 16), dim3(32), 0, stream>>>(z2, e_all, out);
}